// Normalizer_5205500362803
// MI455X (gfx1250) — compile-verified
//
#include <hip/hip_runtime.h>
#include <math.h>

// Normalizer: score[4096][8192] f32, mask[4096][8192] i32 -> gamma f32.
// One workgroup per row; row lives in registers across 20 annealing iterations.
// Row is staged HBM -> LDS with gfx1250 async-to-LDS ops (ASYNCcnt path).

#define SEQ       8192
#define NTHREADS  512
#define EPT       16            // elements per thread
#define NPASS     4             // float4 passes (4 * 512 * 4 = 8192)
#define NWAVES    (NTHREADS / 32)
#define T_ITERS   20

typedef int v4i_t __attribute__((ext_vector_type(4)));

__global__ __launch_bounds__(NTHREADS) void Normalizer_5205500362803_kernel(
    const float* __restrict__ score, const int* __restrict__ mask,
    float* __restrict__ out, int rows)
{
    __shared__ __align__(16) float lds_score[SEQ];
    __shared__ float red[NWAVES];
    __shared__ float bcastv;

    const int row  = blockIdx.x;
    if (row >= rows) return;
    const int tid  = threadIdx.x;
    const int lane = tid & 31;
    const int wave = tid >> 5;

    const float* g_score = score + (size_t)row * SEQ;
    const int*   g_mask  = mask  + (size_t)row * SEQ;
    float*       g_out   = out   + (size_t)row * SEQ;

    // ---- Stage score row into LDS via gfx1250 async copy (ASYNCcnt path) ----
#if defined(__gfx1250__) && __has_builtin(__builtin_amdgcn_global_load_async_to_lds_b128)
    {
        typedef __attribute__((address_space(1))) v4i_t gv4i_t;
        typedef __attribute__((address_space(3))) v4i_t lv4i_t;
        gv4i_t* gs = (gv4i_t*)g_score;     // global source, 16B units
        lv4i_t* ls = (lv4i_t*)lds_score;   // LDS destination, 16B units
#pragma unroll
        for (int p = 0; p < NPASS; ++p)
            __builtin_amdgcn_global_load_async_to_lds_b128(
                gs + p * NTHREADS + tid,
                ls + p * NTHREADS + tid, 0, 0);
    }
#elif defined(__gfx1250__)
    {
        unsigned lds_addr = (unsigned)(size_t)(&lds_score[0]) + (unsigned)(tid * 16);
        const char* gp = (const char*)g_score + tid * 16;
#pragma unroll
        for (int p = 0; p < NPASS; ++p)
            asm volatile("global_load_async_to_lds_b128 %0, %1, off"
                         :: "v"(lds_addr + (unsigned)(p * NTHREADS * 16)),
                            "v"(gp + p * NTHREADS * 16)
                         : "memory");
    }
#else
#pragma unroll
    for (int p = 0; p < NPASS; ++p)
        ((float4*)lds_score)[p * NTHREADS + tid] =
            ((const float4*)g_score)[p * NTHREADS + tid];
#endif

    // Overlap: coalesced mask loads while the async copy is in flight.
    int4 mreg[NPASS];
#pragma unroll
    for (int p = 0; p < NPASS; ++p)
        mreg[p] = ((const int4*)g_mask)[p * NTHREADS + tid];

#if defined(__gfx1250__)
#if __has_builtin(__builtin_amdgcn_s_wait_asynccnt)
    __builtin_amdgcn_s_wait_asynccnt(0);
#else
    asm volatile("s_wait_asynccnt 0x0" ::: "memory");
#endif
#endif
    __syncthreads();

    // ---- Pull row into registers with mask applied; count finite entries ----
    float s[EPT];
    int cnt = 0;
#pragma unroll
    for (int p = 0; p < NPASS; ++p) {
        float4 v = ((const float4*)lds_score)[p * NTHREADS + tid];
        int4   m = mreg[p];
        s[p * 4 + 0] = m.x ? v.x : -INFINITY; cnt += (m.x != 0);
        s[p * 4 + 1] = m.y ? v.y : -INFINITY; cnt += (m.y != 0);
        s[p * 4 + 2] = m.z ? v.z : -INFINITY; cnt += (m.z != 0);
        s[p * 4 + 3] = m.w ? v.w : -INFINITY; cnt += (m.w != 0);
    }

    // ---- k = P * count(finite) via block reduction ----
    float part = (float)cnt;
#pragma unroll
    for (int off = 16; off > 0; off >>= 1) part += __shfl_down(part, off, 32);
    if (lane == 0) red[wave] = part;
    __syncthreads();
    if (tid == 0) {
        float tot = 0.f;
#pragma unroll
        for (int w = 0; w < NWAVES; ++w) tot += red[w];
        bcastv = 0.1f * tot;                 // P = 0.1
    }
    __syncthreads();
    const float k = bcastv;

    // ---- Annealing loop: b = -relu(s+a)  =>  s+b = min(s, -a) ----
    const float LOG2E = 1.4426950408889634f;
    const float LN2   = 0.6931471805599453f;
    float  a   = -INFINITY;                  // makes t=0 equal b=0 case
    double thd = 4.0;                        // THETA0
#pragma unroll 1
    for (int t = 0; t < T_ITERS; ++t) {
        const float th    = (float)(thd > 0.3 ? thd : 0.3);   // max(beta^t*th0, theta)
        thd *= 0.7;                                           // BETA
        const float scale = LOG2E / th;
        const float negA  = -a;

        float acc = 0.f;
#pragma unroll
        for (int i = 0; i < EPT; ++i) {
            float m = fminf(s[i], negA);                       // score + b
            acc += __builtin_amdgcn_exp2f(m * scale);          // v_exp_f32
        }
#pragma unroll
        for (int off = 16; off > 0; off >>= 1) acc += __shfl_down(acc, off, 32);
        if (lane == 0) red[wave] = acc;
        __syncthreads();
        if (tid == 0) {
            float tot = 0.f;
#pragma unroll
            for (int w = 0; w < NWAVES; ++w) tot += red[w];
            float ratio = k / (tot + 1e-20f);
            bcastv = th * (__builtin_amdgcn_logf(ratio) * LN2); // th * ln(ratio)
        }
        __syncthreads();
        a = bcastv;
    }

    // ---- gamma = exp(min(score + a, 0) / THETA), THETA = 0.3 ----
    const float fscale = LOG2E / 0.3f;
#pragma unroll
    for (int p = 0; p < NPASS; ++p) {
        float4 o;
        o.x = __builtin_amdgcn_exp2f(fminf(s[p * 4 + 0] + a, 0.f) * fscale);
        o.y = __builtin_amdgcn_exp2f(fminf(s[p * 4 + 1] + a, 0.f) * fscale);
        o.z = __builtin_amdgcn_exp2f(fminf(s[p * 4 + 2] + a, 0.f) * fscale);
        o.w = __builtin_amdgcn_exp2f(fminf(s[p * 4 + 3] + a, 0.f) * fscale);
        ((float4*)g_out)[p * NTHREADS + tid] = o;
    }
}

extern "C" void kernel_launch(void* const* d_in, const int* in_sizes, int n_in,
                              void* d_out, int out_size, void* d_ws, size_t ws_size,
                              hipStream_t stream) {
    (void)n_in; (void)out_size; (void)d_ws; (void)ws_size;
    const float* score = (const float*)d_in[0];
    const int*   mask  = (const int*)d_in[1];
    float*       out   = (float*)d_out;
    const int rows = in_sizes[0] / SEQ;     // 4096
    Normalizer_5205500362803_kernel<<<dim3(rows), dim3(NTHREADS), 0, stream>>>(
        score, mask, out, rows);
}